// HausdorffLoss_46411416601083
// MI455X (gfx1250) — compile-verified
//
#include <hip/hip_runtime.h>
#include <hip/hip_bf16.h>
#include <float.h>

typedef float v2f __attribute__((ext_vector_type(2)));
typedef float v8f __attribute__((ext_vector_type(8)));

#define B_SZ 4
#define M_SZ 8192
#define K_SZ 8192
#define KCHUNK 2048
#define NT (KCHUNK / 16)          // 128 column tiles per workgroup chunk
#define WAVES_PER_WG 8
#define WMMA_PER_WAVE 4
#define ROWS_PER_WAVE (16 * WMMA_PER_WAVE)            // 64
#define ROWS_PER_WG (WAVES_PER_WG * ROWS_PER_WAVE)    // 512

// ---------------------------------------------------------------------------
// Kernel 1: initialize min arrays (uint bit pattern of +FLT_MAX)
// ---------------------------------------------------------------------------
__global__ void hausdorff_init(unsigned* __restrict__ mins, int n) {
    int i = blockIdx.x * blockDim.x + threadIdx.x;
    if (i < n) mins[i] = 0x7F7FFFFFu;  // bits of FLT_MAX
}

// ---------------------------------------------------------------------------
// Kernel 2: main WMMA distance kernel.
// d2(p,g) = [px,py,|p|^2,1] . [-2gx,-2gy,1,|g|^2]  -> V_WMMA_F32_16X16X4_F32.
// Each wave owns 64 pred rows (4 WMMAs / col tile); WG owns 512 rows x 2048
// gt cols. B operands are pre-swizzled into LDS in exact WMMA lane layout and
// software-pipelined (prefetch t+1 before the reduction chain of t).
// ---------------------------------------------------------------------------
__global__ __launch_bounds__(256) void hausdorff_main(
    const float* __restrict__ pred,   // (B, M, 2)
    const float* __restrict__ gt,     // (B, K, 2)
    unsigned* __restrict__ min_pred,  // (B*M) f32-as-uint squared-dist mins
    unsigned* __restrict__ min_gt)    // (B*K)
{
    // btile[t*32 + lane]: per-lane B operand (2 dwords) for column tile t.
    //   lanes 0-15  : (-2*gx, -2*gy)   = B rows K=0, K=1 for col (lane)
    //   lanes 16-31 : ( 1.0 , |g|^2 )  = B rows K=2, K=3 for col (lane-16)
    __shared__ float2   btile[2 * KCHUNK];   // 32 KB
    __shared__ unsigned colmin[KCHUNK];      //  8 KB

    const int b     = blockIdx.z;
    const int rowWG = blockIdx.x * ROWS_PER_WG;
    const int colWG = blockIdx.y * KCHUNK;
    const int tid   = threadIdx.x;

    // ---- Stage pre-swizzled B operands + init per-WG column mins ----
    for (int j = tid; j < 2 * KCHUNK; j += 256) {
        int t = j >> 5;        // column tile
        int l = j & 31;        // lane slot within tile
        int col = colWG + t * 16 + (l & 15);
        const float* g = &gt[((size_t)b * K_SZ + col) * 2];
        float gx = g[0], gy = g[1];
        btile[j] = (l < 16) ? make_float2(-2.0f * gx, -2.0f * gy)
                            : make_float2(1.0f, gx * gx + gy * gy);
    }
    for (int i = tid; i < KCHUNK; i += 256) colmin[i] = 0x7F7FFFFFu;
    __syncthreads();

    const int wave    = tid >> 5;
    const int lane    = tid & 31;
    const int rowbase = rowWG + wave * ROWS_PER_WAVE;

    // ---- A operands (16x4 f32): lanes 0-15 hold K=0,1 ; lanes 16-31 K=2,3 ----
    v2f A[WMMA_PER_WAVE];
#pragma unroll
    for (int w = 0; w < WMMA_PER_WAVE; ++w) {
        const float* p =
            &pred[((size_t)b * M_SZ + rowbase + w * 16 + (lane & 15)) * 2];
        float px = p[0], py = p[1];
        float pn = px * px + py * py;
        A[w].x = (lane < 16) ? px : pn;    // K=0 | K=2
        A[w].y = (lane < 16) ? py : 1.0f;  // K=1 | K=3
    }

    float rowmin[WMMA_PER_WAVE][8];
#pragma unroll
    for (int w = 0; w < WMMA_PER_WAVE; ++w)
#pragma unroll
        for (int r = 0; r < 8; ++r) rowmin[w][r] = FLT_MAX;

    unsigned* cslot0 = &colmin[lane & 15];  // both half-waves hit same column

    // ---- Main loop: 4 WMMAs + pipelined LDS B-load + 1 LDS col atomic/tile ----
    v2f Bcur;
    {
        float2 b0 = btile[lane];
        Bcur.x = b0.x;
        Bcur.y = b0.y;
    }
#pragma unroll 2
    for (int t = 0; t < NT; ++t) {
        // Prefetch next tile's B operand before this tile's reduction chain,
        // so the ds_load latency hides under ~32 VALU min ops.
        int tn = (t + 1 < NT) ? (t + 1) : 0;
        float2 bn = btile[tn * 32 + lane];

        v8f d[WMMA_PER_WAVE];
#pragma unroll
        for (int w = 0; w < WMMA_PER_WAVE; ++w) {
            v8f c = {};
            d[w] = __builtin_amdgcn_wmma_f32_16x16x4_f32(
                /*neg_a=*/false, A[w], /*neg_b=*/false, Bcur,
                /*c_mod=*/(short)0, c, /*reuse_a=*/false, /*reuse_b=*/false);
        }

        // Column min: fold all 32 per-lane values (seed from first element,
        // no FLT_MAX literal in the min3 tree), then one fire-and-forget
        // LDS atomic from all 32 lanes (2 lanes per column, no branch).
        float cmin = d[0][0];
#pragma unroll
        for (int w = 0; w < WMMA_PER_WAVE; ++w)
#pragma unroll
            for (int r = 0; r < 8; ++r)
                if (w || r) cmin = fminf(cmin, d[w][r]);
        atomicMin(cslot0 + t * 16, __float_as_uint(cmin));

        // Row mins: D layout -> VGPR r, lanes 0-15 = (row base+r, col lane),
        //                              lanes 16-31 = (row base+r+8, col lane-16)
#pragma unroll
        for (int w = 0; w < WMMA_PER_WAVE; ++w)
#pragma unroll
            for (int r = 0; r < 8; ++r)
                rowmin[w][r] = fminf(rowmin[w][r], d[w][r]);

        Bcur.x = bn.x;
        Bcur.y = bn.y;
    }

    // ---- Row-min epilogue: 16 lanes (columns) share each row set ----
#pragma unroll
    for (int w = 0; w < WMMA_PER_WAVE; ++w)
#pragma unroll
        for (int r = 0; r < 8; ++r) {
            float v = rowmin[w][r];
            v = fminf(v, __shfl_xor(v, 1, 32));
            v = fminf(v, __shfl_xor(v, 2, 32));
            v = fminf(v, __shfl_xor(v, 4, 32));
            v = fminf(v, __shfl_xor(v, 8, 32));
            if ((lane & 15) == 0) {
                int row = rowbase + w * 16 + r + ((lane >> 4) << 3);
                atomicMin(&min_pred[(size_t)b * M_SZ + row], __float_as_uint(v));
            }
        }

    // ---- Flush per-WG column mins: one global atomic per column ----
    __syncthreads();
    for (int i = tid; i < KCHUNK; i += 256) {
        atomicMin(&min_gt[(size_t)b * K_SZ + colWG + i], colmin[i]);
    }
}

// ---------------------------------------------------------------------------
// Kernel 3: final reduction -> sqrt + means -> scalar
// ---------------------------------------------------------------------------
__global__ void hausdorff_finalize(const unsigned* __restrict__ min_pred,
                                   const unsigned* __restrict__ min_gt,
                                   float* __restrict__ out) {
    __shared__ float smem[256];
    float accP = 0.0f, accG = 0.0f;
    for (int i = threadIdx.x; i < B_SZ * M_SZ; i += 256)
        accP += sqrtf(__uint_as_float(min_pred[i]));
    for (int i = threadIdx.x; i < B_SZ * K_SZ; i += 256)
        accG += sqrtf(__uint_as_float(min_gt[i]));
    smem[threadIdx.x] = accP * (1.0f / (B_SZ * M_SZ)) + accG * (1.0f / (B_SZ * K_SZ));
    __syncthreads();
    for (int ofs = 128; ofs > 0; ofs >>= 1) {
        if (threadIdx.x < (unsigned)ofs) smem[threadIdx.x] += smem[threadIdx.x + ofs];
        __syncthreads();
    }
    if (threadIdx.x == 0) out[0] = smem[0];
}

// ---------------------------------------------------------------------------
extern "C" void kernel_launch(void* const* d_in, const int* in_sizes, int n_in,
                              void* d_out, int out_size, void* d_ws, size_t ws_size,
                              hipStream_t stream) {
    (void)in_sizes; (void)n_in; (void)out_size; (void)ws_size;

    const float* pred = (const float*)d_in[0];  // (B, M, 2) f32
    const float* gt   = (const float*)d_in[1];  // (B, K, 2) f32
    float* out        = (float*)d_out;          // scalar f32

    unsigned* min_pred = (unsigned*)d_ws;                 // B*M uints
    unsigned* min_gt   = min_pred + (size_t)B_SZ * M_SZ;  // B*K uints

    // 1) init mins to +FLT_MAX
    {
        int n = B_SZ * (M_SZ + K_SZ);
        hausdorff_init<<<(n + 255) / 256, 256, 0, stream>>>(min_pred, n);
    }
    // 2) WMMA distance + min kernel
    {
        dim3 grid(M_SZ / ROWS_PER_WG, K_SZ / KCHUNK, B_SZ);  // (16, 4, 4)
        hausdorff_main<<<grid, 256, 0, stream>>>(pred, gt, min_pred, min_gt);
    }
    // 3) final scalar reduction
    hausdorff_finalize<<<1, 256, 0, stream>>>(min_pred, min_gt, out);
}